// SOLD2_detector_42812234006925
// MI455X (gfx1250) — compile-verified
//
#include <hip/hip_runtime.h>
#include <hip/hip_bf16.h>
#include <math.h>

typedef __attribute__((ext_vector_type(2))) float v2f;
typedef __attribute__((ext_vector_type(8))) float v8f;

#define HH 512
#define WW 512
#define NJ 256
#define HM_N (HH*WW)

// ---- workspace layout (32-bit words) ----
#define WS_H1   0            // 65536 counts (bits>>14 histogram)
#define WS_H2   65536        // 16384 counts (low-14-bit histogram of threshold bin)
#define WS_PART 81920        // 256 deterministic partial sums (float)
#define WS_SCAL 82176        // 16 scalars
#define WS_ZERO_WORDS 82192
#define WS_SUP_BYTE (WS_ZERO_WORDS*4)   // 65536 bytes: suppression mask

#define SC_NVALID 0
#define SC_K      1
#define SC_BINB   2
#define SC_CABOVE 3
#define SC_TVBITS 4
#define SC_CGT    5
#define SC_MEANT  6

// ------------------------------------------------------------------
// 0) zero scratch + line_map
__global__ void k_init(unsigned* ws, int* lm) {
    int idx = blockIdx.x * blockDim.x + threadIdx.x;
    if (idx < WS_ZERO_WORDS) ws[idx] = 0u;
    if (idx < NJ * NJ)       lm[idx] = 0;
}

// 1) coarse histogram over float bit patterns + n_valid count
__global__ void k_hist1(const float* __restrict__ hm, unsigned* ws) {
    int idx = blockIdx.x * blockDim.x + threadIdx.x;
    if (idx >= HM_N) return;
    float v = hm[idx];
    if (v > 0.01f) atomicAdd(&ws[WS_SCAL + SC_NVALID], 1u);
    unsigned bits = __float_as_uint(v);
    unsigned key  = bits >> 14; if (key > 65535u) key = 65535u;
    atomicAdd(&ws[WS_H1 + key], 1u);
}

// 2) find coarse bin containing the k-th largest value
__global__ void k_select1(unsigned* ws) {
    __shared__ unsigned chunkCnt[256];
    __shared__ unsigned sK;
    __shared__ int selChunk;
    __shared__ unsigned cumBefore;
    const int t = threadIdx.x;
    if (t == 0) {
        int nv = (int)ws[WS_SCAL + SC_NVALID];
        int k  = (int)ceilf((float)nv * 0.2f);
        if (k < 1) k = 1;
        ws[WS_SCAL + SC_K] = (unsigned)k;
        sK = (unsigned)k; selChunk = 255; cumBefore = 0;
    }
    __syncthreads();
    const unsigned k = sK;
    unsigned c = 0;
    const int hi = 65535 - t * 256;
    for (int b = 0; b < 256; b++) c += ws[WS_H1 + hi - b];
    chunkCnt[t] = c;
    __syncthreads();
    if (t == 0) {
        unsigned cum = 0;
        for (int q = 0; q < 256; q++) {
            if (cum + chunkCnt[q] >= k) { selChunk = q; cumBefore = cum; break; }
            cum += chunkCnt[q];
        }
    }
    __syncthreads();
    if (t == selChunk) {
        unsigned cum = cumBefore;
        const int h2 = 65535 - t * 256;
        for (int b = 0; b < 256; b++) {
            unsigned cc = ws[WS_H1 + h2 - b];
            if (cum + cc >= k) {
                ws[WS_SCAL + SC_BINB]   = (unsigned)(h2 - b);
                ws[WS_SCAL + SC_CABOVE] = cum;
                break;
            }
            cum += cc;
        }
    }
}

// 3) fine histogram (low 14 bits) of values inside the threshold bin
__global__ void k_hist2(const float* __restrict__ hm, unsigned* ws) {
    int idx = blockIdx.x * blockDim.x + threadIdx.x;
    if (idx >= HM_N) return;
    unsigned binb = ws[WS_SCAL + SC_BINB];
    unsigned bits = __float_as_uint(hm[idx]);
    unsigned key  = bits >> 14; if (key > 65535u) key = 65535u;
    if (key == binb) atomicAdd(&ws[WS_H2 + (bits & 0x3FFFu)], 1u);
}

// 4) exact threshold bit pattern + strictly-greater count
__global__ void k_select2(unsigned* ws) {
    __shared__ unsigned chunkCnt[256];
    __shared__ int selChunk;
    __shared__ unsigned cumBefore;
    const int t = threadIdx.x;
    const unsigned k  = ws[WS_SCAL + SC_K];
    const unsigned cA = ws[WS_SCAL + SC_CABOVE];
    if (t == 0) { selChunk = 255; cumBefore = cA; }
    __syncthreads();
    unsigned c = 0;
    const int hi = 16383 - t * 64;
    for (int b = 0; b < 64; b++) c += ws[WS_H2 + hi - b];
    chunkCnt[t] = c;
    __syncthreads();
    if (t == 0) {
        unsigned cum = cA;
        for (int q = 0; q < 256; q++) {
            if (cum + chunkCnt[q] >= k) { selChunk = q; cumBefore = cum; break; }
            cum += chunkCnt[q];
        }
    }
    __syncthreads();
    if (t == selChunk) {
        unsigned cum = cumBefore;
        const int h2 = 16383 - t * 64;
        unsigned binb = ws[WS_SCAL + SC_BINB];
        for (int b = 0; b < 64; b++) {
            unsigned cc = ws[WS_H2 + h2 - b];
            if (cum + cc >= k) {
                ws[WS_SCAL + SC_TVBITS] = (binb << 14) | (unsigned)(h2 - b);
                ws[WS_SCAL + SC_CGT]    = cum;
                break;
            }
            cum += cc;
        }
    }
}

// 5) deterministic sum of values strictly above threshold (fixed-order tree)
__global__ void k_sumpass(const float* __restrict__ hm, unsigned* ws) {
    __shared__ float sh[256];
    float tv = __uint_as_float(ws[WS_SCAL + SC_TVBITS]);
    int base = blockIdx.x * 1024;
    float s = 0.f;
    for (int i = 0; i < 4; i++) {
        float v = hm[base + threadIdx.x + i * 256];
        s += (v > tv) ? v : 0.f;
    }
    sh[threadIdx.x] = s;
    __syncthreads();
    for (int off = 128; off > 0; off >>= 1) {
        if (threadIdx.x < off) sh[threadIdx.x] += sh[threadIdx.x + off];
        __syncthreads();
    }
    if (threadIdx.x == 0) ((float*)ws)[WS_PART + blockIdx.x] = sh[0];
}

// 6) mean of top-k
__global__ void k_finalize(unsigned* ws) {
    float S = 0.f;
    const float* part = (const float*)ws + WS_PART;
    for (int i = 0; i < 256; i++) S += part[i];
    int   k   = (int)ws[WS_SCAL + SC_K];
    int   cgt = (int)ws[WS_SCAL + SC_CGT];
    float tv  = __uint_as_float(ws[WS_SCAL + SC_TVBITS]);
    float mt  = (S + (float)(k - cgt) * tv) / (float)k;
    ws[WS_SCAL + SC_MEANT] = __float_as_uint(mt);
}

// 7) refined heatmap -> d_out, junction copy -> d_out
__global__ void k_refine(const float* __restrict__ hm, const float* __restrict__ junc,
                         const unsigned* ws, float* outH, float* outJ) {
    int idx = blockIdx.x * blockDim.x + threadIdx.x;
    float mt = __uint_as_float(ws[WS_SCAL + SC_MEANT]);
    if (idx < HM_N) outH[idx] = fminf(fmaxf(hm[idx] / mt, 0.f), 1.f);
    if (idx < NJ * 2) outJ[idx] = junc[idx];
}

// ------------------------------------------------------------------
// 8) candidate suppression: per-i batched [256x2]x[2x256] GEMM on
//    V_WMMA_F32_16X16X4_F32 (K padded 2->4), branchless mask logic on
//    the accumulator (EXEC stays all-ones across the whole WMMA loop).
//    One wave per (i, 16-row j-tile); 16 WMMAs across k-tiles.
__global__ void __launch_bounds__(32) k_suppress(const float* __restrict__ junc,
                                                 unsigned char* __restrict__ sup) {
    const int i    = blockIdx.x;   // start junction
    const int jt   = blockIdx.y;   // j tile (16 rows)
    const int lane = threadIdx.x;  // wave32, EXEC all-ones
    const float jiy = junc[2 * i], jix = junc[2 * i + 1];
    const int inA = (lane < 16) ? 1 : 0;   // lanes 16-31 carry K=2,3 zero pad

    // A fragment: dirv rows (unit direction start->end; NaN when j==i)
    int ja = jt * 16 + (lane & 15);
    float ady = junc[2 * ja] - jiy, adx = junc[2 * ja + 1] - jix;
    float ald = sqrtf(ady * ady + adx * adx);
    v2f a;
    a.x = inA ? (ady / ald) : 0.f;
    a.y = inA ? (adx / ald) : 0.f;

    // D rows this lane owns: M = v (lanes<16) or v+8 (lanes>=16)
    const int mbase = (lane < 16) ? 0 : 8;
    float ld8[8]; int jr8[8];
    for (int v = 0; v < 8; v++) {
        int jr = jt * 16 + mbase + v; jr8[v] = jr;
        float dy = junc[2 * jr] - jiy, dx = junc[2 * jr + 1] - jix;
        ld8[v] = sqrtf(dy * dy + dx * dx);
    }

    int cnt[8]; for (int v = 0; v < 8; v++) cnt[v] = 0;
    const int ksub = lane & 15;

    for (int kt = 0; kt < 16; kt++) {
        int kcol = kt * 16 + ksub;                  // this lane's D column
        float cvy = junc[2 * kcol] - jiy, cvx = junc[2 * kcol + 1] - jix;
        // B fragment: lanes 0-15 = (K=0,K=1) of cand_vecs^T, lanes 16-31 = pad
        v2f b;
        b.x = inA ? cvy : 0.f;
        b.y = inA ? cvx : 0.f;
        v8f c = {};
        c = __builtin_amdgcn_wmma_f32_16x16x4_f32(
                /*neg_a=*/false, a, /*neg_b=*/false, b,
                /*c_mod=*/(short)0, c, /*reuse_a=*/false, /*reuse_b=*/false);
        float cn2 = cvy * cvy + cvx * cvx;
        int notI = (kcol != i) ? 1 : 0;
        #pragma unroll
        for (int v = 0; v < 8; v++) {
            float dd = c[v];                        // dots = cand . unit_dir
            float ld = ld8[v];
            float pp = fmaf(-dd, dd, cn2);          // perp^2 (NaN-safe vs arccos)
            // all-lane branchless predicate: no EXEC writes, pure v_cmp/mask ALU
            int ok = ((int)(dd >= 0.f)) & ((int)(dd <= ld)) &
                     ((int)(pp >= 0.f)) & ((int)(pp <= 9.0f)) &   // NMS_TOL^2
                     notI & ((int)(kcol != jr8[v]));              // diag correction
            cnt[v] += ok;
        }
    }
    // reduce per-row counts within each 16-lane half
    #pragma unroll
    for (int v = 0; v < 8; v++)
        for (int off = 8; off > 0; off >>= 1)
            cnt[v] += __shfl_xor(cnt[v], off);
    if (lane == 0 || lane == 16)
        for (int v = 0; v < 8; v++)
            sup[i * NJ + jr8[v]] = (cnt[v] <= 0) ? 1 : 0;
}

// ------------------------------------------------------------------
// 9) segment sampling + detection; one wave per (i<j) pair.
//    Gather is unconditional (clamped in-bounds, L2-resident heatmap),
//    mask applied as a branchless select.
__global__ void __launch_bounds__(32) k_pairs(const float* __restrict__ junc,
                                              const float* __restrict__ hmR,
                                              const unsigned char* __restrict__ sup,
                                              int* __restrict__ lm) {
    const int i = blockIdx.x, j = blockIdx.y;
    if (i >= j) return;
    const int lane = threadIdx.x;
    float sy = junc[2 * i], sx = junc[2 * i + 1];
    float ey = junc[2 * j], ex = junc[2 * j + 1];
    float dy = sy - ey, dx = sx - ex;
    float segLen = sqrtf(dy * dy + dx * dx);
    float dthr = 0.70710678f + 2.0f * (segLen * (1.0f / 724.07734f)); // sqrt(H^2+W^2)

    float lsum = 0.f; int lcnt = 0;
    for (int rep = 0; rep < 2; rep++) {
        int sidx = lane + rep * 32;                 // 64 samples
        float t  = (float)sidx * (1.0f / 63.0f);
        float ch = fminf(fmaxf(sy * t + ey * (1.f - t), 0.f), (float)(HH - 1));
        float cw = fminf(fmaxf(sx * t + ex * (1.f - t), 0.f), (float)(WW - 1));
        float rh = rintf(ch), rw = rintf(cw);       // round-half-even like jnp.round
        float best = 0.f;                           // feat >= 0 after clip
        #pragma unroll
        for (int dyo = -3; dyo <= 3; dyo++)
            #pragma unroll
            for (int dxo = -3; dxo <= 3; dxo++) {
                if (dyo * dyo + dxo * dxo > 9) continue;   // compile-time prune -> 29
                float py = rh + (float)dyo, px = rw + (float)dxo;
                float d0 = ch - py, d1 = cw - px;
                float pd = sqrtf(d0 * d0 + d1 * d1);
                int hi = (int)fminf(fmaxf(py, 0.f), (float)(HH - 1));
                int wi = (int)fminf(fmaxf(px, 0.f), (float)(WW - 1));
                float f = hmR[hi * WW + wi];               // always in-bounds
                best = fmaxf(best, (pd < dthr) ? f : 0.f); // branchless mask
            }
        lsum += best;
        lcnt += (best > 0.5f) ? 1 : 0;
    }
    for (int off = 16; off > 0; off >>= 1) {
        lsum += __shfl_xor(lsum, off);
        lcnt += __shfl_xor(lcnt, off);
    }
    if (lane == 0) {
        float meanf = lsum * (1.0f / 64.0f);
        bool det = (meanf > 0.5f) &&
                   (((float)lcnt * (1.0f / 64.0f)) >= 0.99f) &&
                   (sup[i * NJ + j] != 0);
        int d32 = det ? 1 : 0;
        lm[i * NJ + j] = d32;
        lm[j * NJ + i] = d32;
    }
}

// ------------------------------------------------------------------
extern "C" void kernel_launch(void* const* d_in, const int* in_sizes, int n_in,
                              void* d_out, int out_size, void* d_ws, size_t ws_size,
                              hipStream_t stream) {
    const float* junc = (const float*)d_in[0];   // [256,2]
    const float* hm   = (const float*)d_in[1];   // [512,512]
    unsigned* ws = (unsigned*)d_ws;
    int*   lm   = (int*)d_out;                   // line_map int32 [256,256]
    float* outJ = (float*)d_out + NJ * NJ;       // junctions [256,2]
    float* outH = (float*)d_out + NJ * NJ + NJ * 2; // refined hm [512,512]
    unsigned char* sup = (unsigned char*)d_ws + WS_SUP_BYTE;

    k_init    <<<(WS_ZERO_WORDS + 255) / 256, 256, 0, stream>>>(ws, lm);
    k_hist1   <<<HM_N / 256, 256, 0, stream>>>(hm, ws);
    k_select1 <<<1, 256, 0, stream>>>(ws);
    k_hist2   <<<HM_N / 256, 256, 0, stream>>>(hm, ws);
    k_select2 <<<1, 256, 0, stream>>>(ws);
    k_sumpass <<<256, 256, 0, stream>>>(hm, ws);
    k_finalize<<<1, 1, 0, stream>>>(ws);
    k_refine  <<<HM_N / 256, 256, 0, stream>>>(hm, junc, ws, outH, outJ);
    k_suppress<<<dim3(NJ, NJ / 16), 32, 0, stream>>>(junc, sup);
    k_pairs   <<<dim3(NJ, NJ), 32, 0, stream>>>(junc, outH, sup, lm);
}